// GETD_HT_43284680409617
// MI455X (gfx1250) — compile-verified
//
#include <hip/hip_runtime.h>
#include <cstdint>

#define N_ENT 50000
#define N_REL 1000
#define D_E 32
#define D_R 32
#define RANK 40
#define BATCH 64
#define EPSV 1e-5f

typedef float v2f __attribute__((ext_vector_type(2)));
typedef float v8f __attribute__((ext_vector_type(8)));

#if defined(__gfx1250__) && __has_builtin(__builtin_amdgcn_global_load_async_to_lds_b32)
#define HAVE_ASYNC_LDS 1
typedef const __attribute__((address_space(1))) float gfloat_t;  // generic->AS1 cast helper
typedef __attribute__((address_space(3))) float lfloat_t;        // generic->AS3 cast helper
typedef __attribute__((address_space(1))) int gint_t;            // builtin's param type
typedef __attribute__((address_space(3))) int lint_t;
#endif

// workspace layout (float offsets)
#define WS_WOUTBN 0        // 64*32   = 2048
#define WS_G      2048     // 32*40*40 = 51200
#define WS_P      53248    // 32*40*1024 = 1310720
#define WS_EBN    1363968  // 4 * 64*32 = 8192 (r_emb, E0, E1, E2)
#define WS_STATS  1372160  // 64*2 = 128

// ---------------------------------------------------------------------------
// G[d,a,b] = sum_c root[d,c] * internal[c,a,b]
__global__ void g_kernel(const float* __restrict__ root,
                         const float* __restrict__ internal,
                         float* __restrict__ G) {
    int t = blockIdx.x * 256 + threadIdx.x;
    if (t >= D_R * RANK * RANK) return;
    int d = t / (RANK * RANK);
    int rem = t % (RANK * RANK);
    int a = rem / RANK, b = rem % RANK;
    float s = 0.f;
    for (int c = 0; c < RANK; ++c)
        s += root[d * RANK + c] * internal[c * RANK * RANK + a * RANK + b];
    G[t] = s;
}

// ---------------------------------------------------------------------------
// Gather + BatchNorm (training-mode, biased stats over batch) for r_emb, e0..e2
__global__ void embed_bn_kernel(const float* __restrict__ E,
                                const float* __restrict__ R,
                                const float* __restrict__ bne_g, const float* __restrict__ bne_b,
                                const float* __restrict__ bnr_g, const float* __restrict__ bnr_b,
                                const int* __restrict__ r_idx, const int* __restrict__ e_idx,
                                float* __restrict__ ws) {
    int t = threadIdx.x;           // 128 threads: 4 tensors x 32 features
    int tensor = t >> 5, f = t & 31;
    float sum = 0.f, sq = 0.f;
    for (int b = 0; b < BATCH; ++b) {
        float v = (tensor == 0) ? R[r_idx[b] * D_R + f]
                                : E[e_idx[(tensor - 1) * BATCH + b] * D_E + f];
        sum += v; sq += v * v;
    }
    float mu = sum * (1.f / BATCH);
    float var = sq * (1.f / BATCH) - mu * mu;
    float g  = (tensor == 0) ? bnr_g[f] : bne_g[f];
    float be = (tensor == 0) ? bnr_b[f] : bne_b[f];
    float rs = rsqrtf(var + EPSV);
    float* dst = ws + WS_EBN + tensor * 2048;
    for (int b = 0; b < BATCH; ++b) {
        float v = (tensor == 0) ? R[r_idx[b] * D_R + f]
                                : E[e_idx[(tensor - 1) * BATCH + b] * D_E + f];
        dst[b * 32 + f] = (v - mu) * rs * g + be;
    }
}

// ---------------------------------------------------------------------------
// Factored HT contraction for W_out (all four miss-domain branches) + BN
__global__ __launch_bounds__(256) void wout_kernel(
        const float* __restrict__ left, const float* __restrict__ right,
        const float* __restrict__ internal, const float* __restrict__ root,
        const float* __restrict__ bnw_g, const float* __restrict__ bnw_b,
        const int* __restrict__ miss_p, float* __restrict__ ws) {
    __shared__ float sRp[BATCH * RANK];
    __shared__ float sSc[BATCH * RANK];
    __shared__ float sM[BATCH * RANK];
    __shared__ float sWout[BATCH * D_E];
    const int tid = threadIdx.x;
    const int miss = miss_p[0];
    const float* remb = ws + WS_EBN;
    const float* E0 = remb + 2048;
    const float* E1 = remb + 4096;
    const float* E2 = remb + 6144;

    // rproj[b,c] = r_emb[b,:] @ root[:,c]
    for (int t = tid; t < BATCH * RANK; t += 256) {
        int bt = t / RANK, c = t % RANK;
        float s = 0.f;
        for (int d = 0; d < D_R; ++d) s += remb[bt * 32 + d] * root[d * RANK + c];
        sRp[t] = s;
    }
    __syncthreads();

    // scalar leaf factors
    for (int t = tid; t < BATCH * RANK; t += 256) {
        int bt = t / RANK, x = t % RANK;
        float s = 0.f;
        if (miss <= 2) {  // Rsc[b,x] = sum_{k,l} right[x,k,l] E1k E2l
            for (int k = 0; k < 32; ++k) {
                float inner = 0.f;
                for (int l = 0; l < 32; ++l) inner += right[x * 1024 + k * 32 + l] * E2[bt * 32 + l];
                s += inner * E1[bt * 32 + k];
            }
        } else {          // Lsc[b,x] = sum_{i,j} left[x,i,j] E0i E1j
            for (int i = 0; i < 32; ++i) {
                float inner = 0.f;
                for (int j = 0; j < 32; ++j) inner += left[x * 1024 + i * 32 + j] * E1[bt * 32 + j];
                s += inner * E0[bt * 32 + i];
            }
        }
        sSc[t] = s;
    }
    __syncthreads();

    // mix through internal core
    for (int t = tid; t < BATCH * RANK; t += 256) {
        int bt = t / RANK, x = t % RANK;
        float s = 0.f;
        if (miss <= 2) {  // Msc[b,a=x]
            for (int c = 0; c < RANK; ++c) {
                float inner = 0.f;
                for (int b = 0; b < RANK; ++b) inner += internal[c * 1600 + x * RANK + b] * sSc[bt * RANK + b];
                s += sRp[bt * RANK + c] * inner;
            }
        } else {          // Nsc[b,bb=x]
            for (int c = 0; c < RANK; ++c) {
                float inner = 0.f;
                for (int a = 0; a < RANK; ++a) inner += internal[c * 1600 + a * RANK + x] * sSc[bt * RANK + a];
                s += sRp[bt * RANK + c] * inner;
            }
        }
        sM[t] = s;
    }
    __syncthreads();

    // project onto output leaf
    for (int t = tid; t < BATCH * D_E; t += 256) {
        int bt = t >> 5, o = t & 31;
        float s = 0.f;
        for (int x = 0; x < RANK; ++x) {
            float inner = 0.f;
            if (miss == 1)      for (int j = 0; j < 32; ++j) inner += left[x * 1024 + o * 32 + j] * E0[bt * 32 + j];
            else if (miss == 2) for (int i = 0; i < 32; ++i) inner += left[x * 1024 + i * 32 + o] * E0[bt * 32 + i];
            else if (miss == 3) for (int l = 0; l < 32; ++l) inner += right[x * 1024 + o * 32 + l] * E2[bt * 32 + l];
            else                for (int k = 0; k < 32; ++k) inner += right[x * 1024 + k * 32 + o] * E2[bt * 32 + k];
            s += sM[bt * RANK + x] * inner;
        }
        sWout[t] = s;
    }
    __syncthreads();

    // BN over batch, write to workspace
    if (tid < 32) {
        int o = tid;
        float sum = 0.f, sq = 0.f;
        for (int b = 0; b < BATCH; ++b) { float v = sWout[b * 32 + o]; sum += v; sq += v * v; }
        float mu = sum * (1.f / BATCH);
        float var = sq * (1.f / BATCH) - mu * mu;
        float rs = rsqrtf(var + EPSV);
        for (int b = 0; b < BATCH; ++b)
            ws[WS_WOUTBN + b * 32 + o] = (sWout[b * 32 + o] - mu) * rs * bnw_g[o] + bnw_b[o];
    }
}

// ---------------------------------------------------------------------------
// P[d,a,kl] = sum_b G[d,a,b] * right[b,kl]
__global__ void pd_kernel(const float* __restrict__ G,
                          const float* __restrict__ right,
                          float* __restrict__ P) {
    int t = blockIdx.x * 256 + threadIdx.x;   // exactly 32*40*1024
    int d = t / (RANK * 1024);
    int rem = t % (RANK * 1024);
    int a = rem >> 10, kl = rem & 1023;
    float s = 0.f;
    for (int b = 0; b < RANK; ++b)
        s += G[d * RANK * RANK + a * RANK + b] * right[b * 1024 + kl];
    P[t] = s;
}

// ---------------------------------------------------------------------------
// Main GEMM: W[d, ij, kl] = Left^T[ij,a] @ P_d[a,kl]   (fp32 WMMA, K=40)
// grid = (16 kl-tiles of 64, 8 ij-supertiles of 128, 32 d), block = 256 (8 waves)
// B tile staged via async global->LDS in fragment-ready float2 pair layout.
__global__ __launch_bounds__(256) void w_wmma_kernel(const float* __restrict__ left,
                                                     const float* __restrict__ P,
                                                     float* __restrict__ Wout) {
    // per k-quad q: half0 at [q*144 + 0, +64), half1 at [q*144 + 80, +64)
    // (half offset 80 float2 == 16 mod 32 -> the two lane-halves hit disjoint banks)
    __shared__ v2f Bq[10 * 144];
    const int tid = threadIdx.x;
    const int wave = tid >> 5, lane = tid & 31;
    const int d = blockIdx.z;
    const int klBase = blockIdx.x * 64;
    const int ijBase = blockIdx.y * 128 + wave * 16;

    {   // stage B tile (40 K-rows x 64 cols) as K-pairs
        const float* Pd = P + (((size_t)d * RANK) << 10) + klBase;
        for (int t = tid; t < 20 * 64; t += 256) {   // 1280/256 = 5, uniform
            int kp = t >> 6, n = t & 63;
            int dst = (kp >> 1) * 144 + (kp & 1) * 80 + n;
#ifdef HAVE_ASYNC_LDS
            float* bsf = (float*)&Bq[0];
            __builtin_amdgcn_global_load_async_to_lds_b32(
                (gint_t*)(gfloat_t*)(Pd + (2 * kp) * 1024 + n),
                (lint_t*)(lfloat_t*)(bsf + 2 * dst), 0, 0);
            __builtin_amdgcn_global_load_async_to_lds_b32(
                (gint_t*)(gfloat_t*)(Pd + (2 * kp + 1) * 1024 + n),
                (lint_t*)(lfloat_t*)(bsf + 2 * dst + 1), 0, 0);
#else
            v2f v;
            v.x = Pd[(2 * kp) * 1024 + n];
            v.y = Pd[(2 * kp + 1) * 1024 + n];
            Bq[dst] = v;
#endif
        }
    }
#ifdef HAVE_ASYNC_LDS
    asm volatile("s_wait_asynccnt 0x0" ::: "memory");
#endif
    __syncthreads();

    v8f zero = {0.f, 0.f, 0.f, 0.f, 0.f, 0.f, 0.f, 0.f};
    v8f acc[4] = {zero, zero, zero, zero};

    // A 16x4 f32 layout: lane<16 -> M=lane, K={k0,k0+1}; lane>=16 -> M=lane-16, K={k0+2,k0+3}
    const int qhalf = lane >> 4;                 // 0 or 1
    const int mrow = ijBase + (lane & 15);
    for (int k0 = 0; k0 < RANK; k0 += 4) {
        const int kA = k0 + qhalf * 2;
        v2f afrag;
        afrag.x = left[kA * 1024 + mrow];        // Left^T[ij, a] = ht_left[a][ij]
        afrag.y = left[(kA + 1) * 1024 + mrow];
        const v2f* brow = &Bq[(k0 >> 2) * 144 + qhalf * 80 + (lane & 15)];
#pragma unroll
        for (int nt = 0; nt < 4; ++nt) {
            v2f bfrag = brow[nt * 16];           // single aligned b64 LDS load
            acc[nt] = __builtin_amdgcn_wmma_f32_16x16x4_f32(
                false, afrag, false, bfrag, (short)0, acc[nt], false, false);
        }
    }

    // C/D layout: VGPR r -> (M=r, N=lane) lanes 0-15 ; (M=r+8, N=lane-16) lanes 16-31
    // r-outer / nt-inner: each row emits 4 stores covering 64 consecutive floats
    const int rowHalf = qhalf * 8;
#pragma unroll
    for (int r = 0; r < 8; ++r) {
        size_t rowOff = ((size_t)d << 20) + (size_t)(ijBase + r + rowHalf) * 1024
                        + klBase + (lane & 15);
#pragma unroll
        for (int nt = 0; nt < 4; ++nt)
            __builtin_nontemporal_store(acc[nt][r], Wout + rowOff + nt * 16);  // write-once stream
    }
}

// ---------------------------------------------------------------------------
// logits[64, n] = Wout_bn[64,32] @ E^T[32, n]   (fp32 WMMA, E tile via LDS)
__global__ __launch_bounds__(256) void logits_wmma_kernel(const float* __restrict__ E,
                                                          const float* __restrict__ ws,
                                                          float* __restrict__ pred) {
    __shared__ float Es[128 * 34];              // [n_local][k], even pad -> aligned b64 frags
    const int tid = threadIdx.x;
    const int nBase = blockIdx.x * 128;
    const float* wb = ws + WS_WOUTBN;

    {   // coalesced load of 128 E rows (clamped at the tail), 16 floats/thread
        int row = tid >> 1, half = (tid & 1) * 16;
        int g = nBase + row; if (g > N_ENT - 1) g = N_ENT - 1;
        const float4* src = (const float4*)(E + (size_t)g * 32 + half);
        float* dst = &Es[row * 34 + half];
#pragma unroll
        for (int q = 0; q < 4; ++q) {
            float4 v = src[q];
            dst[q * 4 + 0] = v.x; dst[q * 4 + 1] = v.y;
            dst[q * 4 + 2] = v.z; dst[q * 4 + 3] = v.w;
        }
    }
    __syncthreads();

    const int wave = tid >> 5, lane = tid & 31;
    const int mtile = (wave & 3) * 16;
    const int nsub = (wave >> 2) * 64;

    v8f zero = {0.f, 0.f, 0.f, 0.f, 0.f, 0.f, 0.f, 0.f};
    v8f acc[4] = {zero, zero, zero, zero};

    const int qhalf = lane >> 4;
    const int mrow = mtile + (lane & 15);
    for (int k0 = 0; k0 < 32; k0 += 4) {
        const int kA = k0 + qhalf * 2;
        v2f afrag = *(const v2f*)&wb[mrow * 32 + kA];     // aligned b64
#pragma unroll
        for (int nt = 0; nt < 4; ++nt) {
            const int nl = nsub + nt * 16 + (lane & 15);
            v2f bfrag = *(const v2f*)&Es[nl * 34 + kA];   // aligned b64 LDS load
            acc[nt] = __builtin_amdgcn_wmma_f32_16x16x4_f32(
                false, afrag, false, bfrag, (short)0, acc[nt], false, false);
        }
    }

    const int rowHalf = qhalf * 8;
#pragma unroll
    for (int r = 0; r < 8; ++r) {
        const size_t rowOff = (size_t)(mtile + r + rowHalf) * N_ENT;
#pragma unroll
        for (int nt = 0; nt < 4; ++nt) {
            const int col = nBase + nsub + nt * 16 + (lane & 15);
            if (col < N_ENT) pred[rowOff + col] = acc[nt][r];
        }
    }
}

// ---------------------------------------------------------------------------
// per-row max and sum(exp(x-max)) over 50000 logits
__global__ __launch_bounds__(256) void rowstat_kernel(const float* __restrict__ pred,
                                                      float* __restrict__ ws) {
    __shared__ float red[256];
    __shared__ float smax;
    const int row = blockIdx.x, tid = threadIdx.x;
    const float* p = pred + (size_t)row * N_ENT;
    float m = -3.4e38f;
    for (int c = tid; c < N_ENT; c += 256) m = fmaxf(m, p[c]);
    red[tid] = m;
    __syncthreads();
    for (int s = 128; s > 0; s >>= 1) {
        if (tid < s) red[tid] = fmaxf(red[tid], red[tid + s]);
        __syncthreads();
    }
    if (tid == 0) smax = red[0];
    __syncthreads();
    const float mx = smax;
    float sum = 0.f;
    for (int c = tid; c < N_ENT; c += 256) sum += expf(p[c] - mx);
    red[tid] = sum;
    __syncthreads();
    for (int s = 128; s > 0; s >>= 1) {
        if (tid < s) red[tid] += red[tid + s];
        __syncthreads();
    }
    if (tid == 0) { ws[WS_STATS + row * 2] = mx; ws[WS_STATS + row * 2 + 1] = red[0]; }
}

__global__ void softmax_kernel(float* __restrict__ pred, const float* __restrict__ ws) {
    size_t idx = (size_t)blockIdx.x * 256 + threadIdx.x;
    if (idx >= (size_t)BATCH * N_ENT) return;
    int row = (int)(idx / N_ENT);
    float mx = ws[WS_STATS + row * 2];
    float sm = ws[WS_STATS + row * 2 + 1];
    pred[idx] = expf(pred[idx] - mx) / sm;
}

// ---------------------------------------------------------------------------
extern "C" void kernel_launch(void* const* d_in, const int* in_sizes, int n_in,
                              void* d_out, int out_size, void* d_ws, size_t ws_size,
                              hipStream_t stream) {
    (void)in_sizes; (void)n_in; (void)out_size; (void)ws_size;
    const float* E     = (const float*)d_in[0];
    const float* R     = (const float*)d_in[1];
    const float* hl    = (const float*)d_in[2];
    const float* hr    = (const float*)d_in[3];
    const float* hi    = (const float*)d_in[4];
    const float* hroot = (const float*)d_in[5];
    const float* bne_g = (const float*)d_in[6];
    const float* bne_b = (const float*)d_in[7];
    const float* bnr_g = (const float*)d_in[8];
    const float* bnr_b = (const float*)d_in[9];
    const float* bnw_g = (const float*)d_in[10];
    const float* bnw_b = (const float*)d_in[11];
    const int* r_idx   = (const int*)d_in[12];
    const int* e_idx   = (const int*)d_in[13];
    const int* miss    = (const int*)d_in[14];

    float* ws   = (float*)d_ws;
    float* pred = (float*)d_out;                       // [64, 50000]
    float* W    = pred + (size_t)BATCH * N_ENT;        // [32,32,32,32,32]

    g_kernel<<<(D_R * RANK * RANK + 255) / 256, 256, 0, stream>>>(hroot, hi, ws + WS_G);
    embed_bn_kernel<<<1, 128, 0, stream>>>(E, R, bne_g, bne_b, bnr_g, bnr_b, r_idx, e_idx, ws);
    wout_kernel<<<1, 256, 0, stream>>>(hl, hr, hi, hroot, bnw_g, bnw_b, miss, ws);
    pd_kernel<<<(D_R * RANK * 1024) / 256, 256, 0, stream>>>(ws + WS_G, hr, ws + WS_P);
    w_wmma_kernel<<<dim3(16, 8, 32), 256, 0, stream>>>(hl, ws + WS_P, W);
    logits_wmma_kernel<<<(N_ENT + 127) / 128, 256, 0, stream>>>(E, ws, pred);
    rowstat_kernel<<<BATCH, 256, 0, stream>>>(pred, ws);
    softmax_kernel<<<(BATCH * N_ENT + 255) / 256, 256, 0, stream>>>(pred, ws);
}